// SparseLinearLayer_24086176596651
// MI455X (gfx1250) — compile-verified
//
#include <hip/hip_runtime.h>

#define N_GENES        2048
#define BATCH          2048
#define GENES_PER_TILE 16
#define N_TILES        (N_GENES / GENES_PER_TILE)   // 128
#define MAX_K          384                          // 16 genes * max block 24
#define BLOCK_THREADS  256
#define ROWS_PER_WAVE  32                           // two 16x16 C tiles per wave
#define ROWS_PER_BLOCK 256                          // 8 waves * 32 batch rows
#define N_BATCH_CHUNKS (BATCH / ROWS_PER_BLOCK)     // 8

typedef float v2f __attribute__((ext_vector_type(2)));
typedef float v8f __attribute__((ext_vector_type(8)));

// ---------------------------------------------------------------------------
// Kernel A: tile boundaries (start row of genes 0,16,...,2048).
// Block sizes in [8,24] => start[G] in [8G, min(24G, input_dim)], and the
// one-hot column of mask row j lies in [j/24-1, j/8]. Binary search over rows
// with a cooperative window scan => ~10 MB of mask traffic instead of 256 MB.
// ---------------------------------------------------------------------------
__global__ __launch_bounds__(256) void gene_tile_boundaries(
    const float* __restrict__ mask, int input_dim, int* __restrict__ bnd) {
  const int t = blockIdx.x;                       // 0..N_TILES
  if (t == 0)        { if (threadIdx.x == 0) bnd[0] = 0;               return; }
  if (t == N_TILES)  { if (threadIdx.x == 0) bnd[N_TILES] = input_dim; return; }

  const int G = t * GENES_PER_TILE;
  __shared__ int s_pos;
  int lo = 8 * G;
  int hi = 24 * G; if (hi > input_dim) hi = input_dim;

  while (lo < hi) {                               // uniform across block
    const int mid = (lo + hi) >> 1;
    int glo = mid / 24; glo = (glo > 0) ? glo - 1 : 0;
    int ghi = mid / 8;  if (ghi > N_GENES - 1) ghi = N_GENES - 1;
    if (threadIdx.x == 0) s_pos = -1;
    __syncthreads();
    const float* row = mask + (long long)mid * N_GENES;
    for (int g = glo + (int)threadIdx.x; g <= ghi; g += 256)
      if (row[g] != 0.0f) s_pos = g;              // exactly one hit per row
    __syncthreads();
    const int pos = s_pos;
    __syncthreads();                              // protect s_pos reuse
    if (pos >= G) hi = mid; else lo = mid + 1;
  }
  if (threadIdx.x == 0) bnd[t] = lo;
}

// ---------------------------------------------------------------------------
// Kernel B: per gene-tile band GEMM with V_WMMA_F32_16X16X4_F32.
// grid = (128 tiles, 8 batch chunks); 256 threads = 8 waves; each wave owns
// TWO 16(batch) x 16(gene) tiles sharing the same B (halves DS traffic).
// K loop unrolled 4x: 8 global b64 loads + 4 ds_2addr issued before 8 WMMAs
// so each wave keeps ~2KB of x in flight (MLP for the 23.3 TB/s stream).
// ---------------------------------------------------------------------------
__global__ __launch_bounds__(BLOCK_THREADS) void sparse_linear_wmma(
    const float* __restrict__ x, const float* __restrict__ w,
    const float* __restrict__ bias, const int* __restrict__ bnd,
    float* __restrict__ out, int input_dim) {
  __shared__ float s_w[MAX_K * GENES_PER_TILE];   // 24 KB

  const int t    = blockIdx.x;
  const int g0   = t * GENES_PER_TILE;
  const int j0   = bnd[t];
  const int Kt   = bnd[t + 1] - j0;               // 128..384
  const int Kpad = (Kt + 3) & ~3;

  // Stage w tile: s_w[k*16 + n] = w[j0+k, g0+n]  (64B contiguous per k-row)
  for (int idx = threadIdx.x; idx < Kt * GENES_PER_TILE; idx += BLOCK_THREADS) {
    const int k = idx >> 4, n = idx & 15;
    s_w[idx] = w[(long long)(j0 + k) * N_GENES + (g0 + n)];
  }
  for (int idx = Kt * GENES_PER_TILE + (int)threadIdx.x;
       idx < Kpad * GENES_PER_TILE; idx += BLOCK_THREADS)
    s_w[idx] = 0.0f;                              // zero-pad K tail
  __syncthreads();

  const int lane  = threadIdx.x & 31;
  const int wv    = threadIdx.x >> 5;
  const int b0    = blockIdx.y * ROWS_PER_BLOCK + wv * ROWS_PER_WAVE;
  const int n     = lane & 15;                    // gene col / row-in-tile
  const int khalf = (lane >> 4) << 1;             // 0 (lanes 0-15) or 2 (16-31)

  // A layout: lanes 0-15 row M=lane hold K,K+1; lanes 16-31 same rows K+2,K+3
  const float* xr0 = x + (long long)(b0 + n)      * input_dim + j0;
  const float* xr1 = x + (long long)(b0 + 16 + n) * input_dim + j0;

#define LDA(dst, src, kofs)                                        \
  do { (dst).x = (src)[(kofs) + khalf];                            \
       (dst).y = (src)[(kofs) + khalf + 1]; } while (0)
#define LDB(dst, kofs)                                             \
  do { (dst).x = s_w[((kofs) + khalf)     * GENES_PER_TILE + n];   \
       (dst).y = s_w[((kofs) + khalf + 1) * GENES_PER_TILE + n]; } while (0)
#define MMA(acc, a, b)                                             \
  (acc) = __builtin_amdgcn_wmma_f32_16x16x4_f32(false, (a), false, (b), \
                                                (short)0, (acc), false, false)

  v8f acc0 = {}, acc1 = {};
  const int K_main = Kt & ~3;
  int kk = 0;
  for (; kk + 16 <= K_main; kk += 16) {           // 4x unrolled body
    v2f a0, a1, a2, a3, c0, c1, c2, c3, e0, e1, e2, e3;
    LDA(a0, xr0, kk);      LDA(c0, xr1, kk);
    LDA(a1, xr0, kk + 4);  LDA(c1, xr1, kk + 4);
    LDA(a2, xr0, kk + 8);  LDA(c2, xr1, kk + 8);
    LDA(a3, xr0, kk + 12); LDA(c3, xr1, kk + 12);
    LDB(e0, kk); LDB(e1, kk + 4); LDB(e2, kk + 8); LDB(e3, kk + 12);
    MMA(acc0, a0, e0); MMA(acc1, c0, e0);
    MMA(acc0, a1, e1); MMA(acc1, c1, e1);
    MMA(acc0, a2, e2); MMA(acc1, c2, e2);
    MMA(acc0, a3, e3); MMA(acc1, c3, e3);
  }
  for (; kk < K_main; kk += 4) {                  // K%16 remainder
    v2f a, c, e;
    LDA(a, xr0, kk); LDA(c, xr1, kk); LDB(e, kk);
    MMA(acc0, a, e); MMA(acc1, c, e);
  }
  if (K_main < Kt) {                              // K%4 tail, EXEC stays full
    v2f a, c, e;
    a.x = (kk + khalf     < Kt) ? xr0[kk + khalf]     : 0.0f;
    a.y = (kk + khalf + 1 < Kt) ? xr0[kk + khalf + 1] : 0.0f;
    c.x = (kk + khalf     < Kt) ? xr1[kk + khalf]     : 0.0f;
    c.y = (kk + khalf + 1 < Kt) ? xr1[kk + khalf + 1] : 0.0f;
    LDB(e, kk);                                   // zero-padded rows
    MMA(acc0, a, e); MMA(acc1, c, e);
  }
#undef LDA
#undef LDB
#undef MMA

  // C/D layout: VGPR v -> M = v (+8 for lanes 16-31), N = lane & 15
  const float bv   = bias[g0 + n];
  const int   brow = b0 + ((lane >> 4) << 3);
#pragma unroll
  for (int v = 0; v < 8; ++v) {
    out[(long long)(brow + v)      * N_GENES + (g0 + n)] = acc0[v] + bv;
    out[(long long)(brow + 16 + v) * N_GENES + (g0 + n)] = acc1[v] + bv;
  }
}

// ---------------------------------------------------------------------------
extern "C" void kernel_launch(void* const* d_in, const int* in_sizes, int n_in,
                              void* d_out, int out_size, void* d_ws, size_t ws_size,
                              hipStream_t stream) {
  const float* x    = (const float*)d_in[0];
  const float* w    = (const float*)d_in[1];
  const float* bias = (const float*)d_in[2];
  const float* mask = (const float*)d_in[3];
  float* out = (float*)d_out;

  const int input_dim = in_sizes[0] / BATCH;      // ~32768
  int* bnd = (int*)d_ws;                          // 129 ints

  gene_tile_boundaries<<<N_TILES + 1, 256, 0, stream>>>(mask, input_dim, bnd);

  dim3 grid(N_TILES, N_BATCH_CHUNKS);
  sparse_linear_wmma<<<grid, BLOCK_THREADS, 0, stream>>>(x, w, bias, bnd, out, input_dim);
}